// FDWConv3d_35837207118546
// MI455X (gfx1250) — compile-verified
//
#include <hip/hip_runtime.h>

typedef _Float16 v16h __attribute__((ext_vector_type(16)));
typedef _Float16 v8h  __attribute__((ext_vector_type(8)));
typedef _Float16 h2   __attribute__((ext_vector_type(2)));
typedef float    v8f  __attribute__((ext_vector_type(8)));

#define IN_CH   32
#define OUT_CH  64
#define DHW     262144      // 64*64*64
#define WROW    32          // halves per [oc] row in weight buffer
#define XPITCH  40          // padded halves per LDS x row (80B, 16B aligned, bank-friendly)
#define XROWS   792         // 3(dz) * 4(y) * 66(x)
#define XBYTES  (XROWS * XPITCH * 2)   // 63360 B dynamic LDS

// ---------------------------------------------------------------------------
// Kernel 1: global average pool.  One block per (b, c) channel (64 blocks).
// ---------------------------------------------------------------------------
__global__ __launch_bounds__(256) void fdw_pool(const float* __restrict__ x,
                                                float* __restrict__ pooled) {
    const int bc = blockIdx.x;
    const float4* p4 = (const float4*)(x + (size_t)bc * DHW);
    float s = 0.f;
    for (int i = threadIdx.x; i < DHW / 4; i += 256) {
        float4 v = p4[i];
        s += v.x + v.y + v.z + v.w;
    }
    __shared__ float red[256];
    red[threadIdx.x] = s;
    __syncthreads();
    for (int off = 128; off > 0; off >>= 1) {
        if (threadIdx.x < off) red[threadIdx.x] += red[threadIdx.x + off];
        __syncthreads();
    }
    if (threadIdx.x == 0) pooled[bc] = red[0] * (1.f / (float)DHW);
}

// ---------------------------------------------------------------------------
// Kernel 2: attention MLP -> softmax -> band gains g[b][band] (band = #nonzero
// freq components; quantile masks collapse to g = {a0, a0, a1+a2, a2+a3}).
// ---------------------------------------------------------------------------
__global__ void fdw_att(const float* __restrict__ pooled,
                        const float* __restrict__ w1, const float* __restrict__ b1,
                        const float* __restrict__ w2, const float* __restrict__ b2,
                        float* __restrict__ g) {
    const int b = threadIdx.x;
    if (b >= 2) return;
    float h[8];
    for (int j = 0; j < 8; ++j) {
        float s = b1[j];
        for (int c = 0; c < IN_CH; ++c) s += w1[j * IN_CH + c] * pooled[b * IN_CH + c];
        h[j] = s > 0.f ? s : 0.f;
    }
    float l[4], mx = -1e30f;
    for (int k = 0; k < 4; ++k) {
        float s = b2[k];
        for (int j = 0; j < 8; ++j) s += w2[k * 8 + j] * h[j];
        l[k] = s;
        mx = fmaxf(mx, s);
    }
    float e[4], sum = 0.f;
    for (int k = 0; k < 4; ++k) { e[k] = __expf(l[k] - mx); sum += e[k]; }
    const float inv = 1.f / sum;
    const float a0 = e[0] * inv, a1 = e[1] * inv, a2 = e[2] * inv, a3 = e[3] * inv;
    g[b * 4 + 0] = a0;
    g[b * 4 + 1] = a0;
    g[b * 4 + 2] = a1 + a2;
    g[b * 4 + 3] = a2 + a3;
}

// ---------------------------------------------------------------------------
// Kernel 3: build per-batch effective f16 weights via 27-point inverse DFT.
// W[b][tap][oc][ic] = (1/27) * sum_f g_b[band(f)] * (Re*cos - Im*sin).
// One thread per (b, oc, ic) -> 4096 threads.
// ---------------------------------------------------------------------------
__global__ __launch_bounds__(256) void fdw_build_weights(
        const float* __restrict__ fp, const float* __restrict__ g,
        _Float16* __restrict__ Wg) {
    const int t  = blockIdx.x * 256 + threadIdx.x;   // [0, 4096)
    const int b  = t >> 11;
    const int oi = t & 2047;                          // oc*32 + ic
    const float gs0 = g[b * 4 + 0], gs1 = g[b * 4 + 1],
                gs2 = g[b * 4 + 2], gs3 = g[b * 4 + 3];
    float s[27];
#pragma unroll
    for (int n = 0; n < 27; ++n) s[n] = 0.f;

    const float* p = fp + (size_t)oi * 54;            // 27 freqs x (re, im)
    for (int f = 0; f < 27; ++f) {
        const int kd = f / 9, kr = f - kd * 9, kh = kr / 3, kw = kr - kh * 3;
        const int c = (kd != 0) + (kh != 0) + (kw != 0);
        const float gv = (c == 0) ? gs0 : (c == 1) ? gs1 : (c == 2) ? gs2 : gs3;
        const float re = p[f * 2] * gv;
        const float im = p[f * 2 + 1] * gv;
        // exp(2*pi*i*m/3): cos = {1,-1/2,-1/2}, sin = {0, +s3, -s3}
        const float t0 = re;
        const float t1 = -0.5f * re - 0.8660254037844386f * im;
        const float t2 = -0.5f * re + 0.8660254037844386f * im;
#pragma unroll
        for (int n = 0; n < 27; ++n) {
            const int nd = n / 9, nr = n - nd * 9, nh = nr / 3, nw = nr - nh * 3;
            const int m = (kd * nd + kh * nh + kw * nw) % 3;
            s[n] += (m == 0) ? t0 : ((m == 1) ? t1 : t2);
        }
    }
    _Float16* o = Wg + (size_t)b * 55296 + oi;        // [b][tap][oc*32+ic]
#pragma unroll
    for (int n = 0; n < 27; ++n)
        o[(size_t)n * 2048] = (_Float16)(s[n] * (1.f / 27.f));
}

// ---------------------------------------------------------------------------
// Kernel 4: implicit-GEMM 3D conv with WMMA f32_16x16x32_f16.
// Block = (b, d0, h0..h0+1): 64 oc x 2 h x 64 w outputs, 8 waves.
// Wave (og, ty): oc [og*16, +16), h = h0+ty, w 0..63 (4 WMMA N-tiles).
// K-loop: 27 taps, K = in_ch = 32 per WMMA.  Accumulators are kept stationary
// via tied inline-asm WMMA operands (no acc moves, no per-WMMA hazard NOPs:
// back-to-back WMMAs on disjoint D with shared A have no D->A/B RAW).
// ---------------------------------------------------------------------------
__device__ __forceinline__ v16h ldsB(const _Float16* p) {
    const v8h lo = *(const v8h*)p;         // K 0-7  (hi lanes: 16-23)
    const v8h hi = *(const v8h*)(p + 8);   // K 8-15 (hi lanes: 24-31)
    return __builtin_shufflevector(lo, hi,
            0, 1, 2, 3, 4, 5, 6, 7, 8, 9, 10, 11, 12, 13, 14, 15);
}

__global__ __launch_bounds__(256) void fdw_conv(
        const float* __restrict__ x, const _Float16* __restrict__ Wg,
        const float* __restrict__ bias, float* __restrict__ out) {
    extern __shared__ char smem[];
    _Float16* xl = (_Float16*)smem;                   // [792][XPITCH] halves

    const int tid = threadIdx.x;
    const int h0  = blockIdx.x * 2;
    const int d0  = blockIdx.y;
    const int b   = blockIdx.z;

    // ---- stage haloed x tile (3dz x 4y x 66x x 32ic), f32 -> f16, [pos][ic] ----
    const float* xb = x + (size_t)b * IN_CH * DHW;
#pragma unroll
    for (int k = 0; k < 4; ++k) {
        const int pos = tid + (k << 8);
        if (pos < XROWS) {
            const int r  = pos / 66;
            const int xx = pos - r * 66;
            const int dz = r >> 2, y = r & 3;
            const int gd = d0 + dz - 1, gh = h0 + y - 1, gw = xx - 1;
            const bool ok = ((unsigned)gd < 64u) && ((unsigned)gh < 64u) && ((unsigned)gw < 64u);
            const float* src = xb + ((long long)gd * 4096 + (long long)gh * 64 + gw);
            _Float16* dst = xl + pos * XPITCH;
#pragma unroll 4
            for (int i = 0; i < IN_CH; i += 2) {       // packed ds_store_b32
                const float v0 = ok ? src[(size_t)i * DHW] : 0.f;
                const float v1 = ok ? src[(size_t)(i + 1) * DHW] : 0.f;
                h2 p; p[0] = (_Float16)v0; p[1] = (_Float16)v1;
                *(h2*)(dst + i) = p;
            }
        }
    }
    __syncthreads();

    const int  lane = tid & 31;
    const int  wid  = tid >> 5;
    const int  og   = wid & 3;         // out-channel group (16 oc)
    const int  ty   = wid >> 2;        // h row within block (0..1)
    const int  m    = lane & 15;
    const bool hiL  = lane >= 16;

    // A fragment source: weights are L2-resident, read as 2x global_load_b128.
    // 16-bit A layout: lanes 0-15 hold K{0-7,16-23}, lanes 16-31 hold K{8-15,24-31}.
    const _Float16* wbase = Wg + (size_t)b * 55296
                          + (size_t)(og * 16 + m) * WROW + (hiL ? 8 : 0);

    v8f acc0 = {}, acc1 = {}, acc2 = {}, acc3 = {};

    for (int tap = 0; tap < 27; ++tap) {
        const int dz = tap / 9, tr = tap - dz * 9, dy = tr / 3, dx = tr - dy * 3;

        const _Float16* ap = wbase + (size_t)tap * 2048;
        const v8h alo = *(const v8h*)ap;              // K 0-7   (hi lanes: 8-15)
        const v8h ahi = *(const v8h*)(ap + 16);       // K 16-23 (hi lanes: 24-31)
        const v16h a = __builtin_shufflevector(alo, ahi,
                0, 1, 2, 3, 4, 5, 6, 7, 8, 9, 10, 11, 12, 13, 14, 15);

        // B as loaded here: lane = N column, K packed 2/VGPR; hi lanes K 16-31 (contiguous order).
        const _Float16* brow = xl
            + (size_t)((dz * 4 + ty + dy) * 66 + dx + m) * XPITCH + (hiL ? 16 : 0);
        const v16h b0 = ldsB(brow);
        const v16h b1 = ldsB(brow + 16 * XPITCH);
        const v16h b2 = ldsB(brow + 32 * XPITCH);
        const v16h b3 = ldsB(brow + 48 * XPITCH);

        asm volatile("v_wmma_f32_16x16x32_f16 %0, %1, %2, %0"
                     : "+v"(acc0) : "v"(a), "v"(b0));
        asm volatile("v_wmma_f32_16x16x32_f16 %0, %1, %2, %0"
                     : "+v"(acc1) : "v"(a), "v"(b1));
        asm volatile("v_wmma_f32_16x16x32_f16 %0, %1, %2, %0"
                     : "+v"(acc2) : "v"(a), "v"(b2));
        asm volatile("v_wmma_f32_16x16x32_f16 %0, %1, %2, %0"
                     : "+v"(acc3) : "v"(a), "v"(b3));
    }

    // WMMA(f16) -> VALU hazard: 5 co-exec NOPs before VALU touches D.
    asm volatile("v_nop\n\tv_nop\n\tv_nop\n\tv_nop\n\tv_nop" ::);

    // ---- write back: VGPR r -> M=r (lanes 0-15) / M=r+8 (lanes 16-31) ----
    const int h = h0 + ty;
    const int oc0 = og * 16 + (hiL ? 8 : 0);
    float bv[8];
#pragma unroll
    for (int r = 0; r < 8; ++r) bv[r] = bias[oc0 + r];
    const size_t obase = ((size_t)(b * OUT_CH + oc0) * 64 + d0) * 4096 + (size_t)h * 64;
    const v8f accs[4] = {acc0, acc1, acc2, acc3};
#pragma unroll
    for (int j = 0; j < 4; ++j) {
        const int wc = j * 16 + m;
#pragma unroll
        for (int r = 0; r < 8; ++r)
            __builtin_nontemporal_store(accs[j][r] + bv[r],
                                        out + obase + (size_t)r * DHW + wc);
    }
}

// ---------------------------------------------------------------------------
extern "C" void kernel_launch(void* const* d_in, const int* in_sizes, int n_in,
                              void* d_out, int out_size, void* d_ws, size_t ws_size,
                              hipStream_t stream) {
    (void)in_sizes; (void)n_in; (void)out_size; (void)ws_size;
    const float* x    = (const float*)d_in[0];   // (2,32,64,64,64)
    const float* fp   = (const float*)d_in[1];   // (64,32,3,3,3,2)
    const float* w1   = (const float*)d_in[2];   // (8,32)
    const float* b1   = (const float*)d_in[3];   // (8,)
    const float* w2   = (const float*)d_in[4];   // (4,8)
    const float* b2   = (const float*)d_in[5];   // (4,)
    const float* bias = (const float*)d_in[6];   // (64,)
    float* out = (float*)d_out;                  // (2,64,64,64,64)

    // workspace: pooled[64] f32 | g[8] f32 | W f16 [2][27][64][32]  (~222 KB)
    float*    pooled = (float*)d_ws;
    float*    g      = pooled + 64;
    _Float16* Wg     = (_Float16*)((char*)d_ws + 512);

    fdw_pool<<<dim3(2 * IN_CH), dim3(256), 0, stream>>>(x, pooled);
    fdw_att<<<dim3(1), dim3(32), 0, stream>>>(pooled, w1, b1, w2, b2, g);
    fdw_build_weights<<<dim3(16), dim3(256), 0, stream>>>(fp, g, Wg);
    fdw_conv<<<dim3(32, 64, 2), dim3(256), XBYTES, stream>>>(x, Wg, bias, out);
}